// ShiftingLayer_63247688401341
// MI455X (gfx1250) — compile-verified
//
#include <hip/hip_runtime.h>
#include <hip/hip_bf16.h>

// D = A(16x4,f32) * B(4x16,f32) + C(16x16,f32), wave32.
typedef __attribute__((ext_vector_type(2))) float v2f;
typedef __attribute__((ext_vector_type(8))) float v8f;

namespace {
constexpr int kB = 16;
constexpr int kS = 4096;
constexpr int kC = 128;
constexpr int kT = 2048;
// Gaussian window: exp(-4*d^2) == 0 in fp32 for |d| >= ~5.1.
// Halo of 16 rows on each side of a 16-row t-tile covers |shift| <= ~10.9.
constexpr int kHalo   = 16;
constexpr int kKSteps = (16 + 2 * kHalo) / 4;  // 12 WMMA K-steps of 4
constexpr float kInvWidth = 4.0f;              // 1 / 0.25
}  // namespace

__global__ __launch_bounds__(256)
void ShiftingLayer_63247688401341_kernel(const float* __restrict__ inp,  // (B,S,C)
                                         const float* __restrict__ sw,   // (B,T)
                                         float* __restrict__ out)        // (B,T,C)
{
    const int t0   = blockIdx.x * 16;        // t-tile base
    const int b    = blockIdx.y;             // batch
    const int wave = threadIdx.x >> 5;       // 0..7 -> 16-wide C slice
    const int lane = threadIdx.x & 31;
    const int half = lane >> 4;              // 0: lanes 0-15, 1: lanes 16-31
    const int l16  = lane & 15;

    const int c0 = wave * 16;

    // A-matrix row m owned by this lane (per 32-bit 16x4 A layout: M = lane%16).
    const int   m     = l16;
    const float shift = sw[b * kT + t0 + m];
    const float tpos  = (float)(t0 + m);

    const float* __restrict__ inB = inp + (size_t)b * kS * kC;

    v8f acc = {};
    #pragma unroll
    for (int kstep = 0; kstep < kKSteps; ++kstep) {
        // K index for this lane's VGPR0 element (VGPR0: K=2*half, VGPR1: K=2*half+1).
        const int s0 = t0 - kHalo + kstep * 4 + half * 2;
        const int s1 = s0 + 1;

        // ---- A: w(m, s) = exp(-4 * (shift + s - t)^2) ----
        const float d0 = shift + (float)s0 - tpos;
        const float d1 = d0 + 1.0f;
        v2f a;
        a.x = __expf(-kInvWidth * d0 * d0);
        a.y = __expf(-kInvWidth * d1 * d1);
        // Zero the weight when the s row is out of range (mirrors reference's
        // implicit zero contribution; keeps everything branch-free).
        a.x = (s0 >= 0 && s0 < kS) ? a.x : 0.0f;
        a.y = (s1 >= 0 && s1 < kS) ? a.y : 0.0f;

        // ---- B: input rows s0,s1 at column c0 + (lane%16) ----
        // Clamp address, select value: unconditional loads + v_cndmask, no EXEC
        // divergence around the WMMA.
        const int s0c = s0 < 0 ? 0 : (s0 >= kS ? kS - 1 : s0);
        const int s1c = s1 < 0 ? 0 : (s1 >= kS ? kS - 1 : s1);
        float bx = inB[(size_t)s0c * kC + c0 + l16];
        float by = inB[(size_t)s1c * kC + c0 + l16];
        v2f bm;
        bm.x = (s0 >= 0 && s0 < kS) ? bx : 0.0f;
        bm.y = (s1 >= 0 && s1 < kS) ? by : 0.0f;

        // (neg_a, A, neg_b, B, c_mod, C, reuse_a, reuse_b)
        acc = __builtin_amdgcn_wmma_f32_16x16x4_f32(
            false, a, false, bm, (short)0, acc, false, false);
    }

    // ---- Store D: 16x16 f32 C/D layout: VGPR r, half h -> row M = r + 8*h ----
    float* __restrict__ outB = out + ((size_t)b * kT + t0) * kC;
    #pragma unroll
    for (int r = 0; r < 8; ++r) {
        outB[(size_t)(r + 8 * half) * kC + c0 + l16] = acc[r];
    }
}

extern "C" void kernel_launch(void* const* d_in, const int* in_sizes, int n_in,
                              void* d_out, int out_size, void* d_ws, size_t ws_size,
                              hipStream_t stream) {
    (void)in_sizes; (void)n_in; (void)d_ws; (void)ws_size; (void)out_size;
    const float* inp = (const float*)d_in[0];   // (16, 4096, 128) fp32
    const float* sw  = (const float*)d_in[1];   // (16, 2048) fp32
    float*       out = (float*)d_out;           // (16, 2048, 128) fp32

    dim3 grid(kT / 16, kB);   // (128, 16) tiles
    dim3 block(256);          // 8 wave32 waves: one 16-wide C slice each
    ShiftingLayer_63247688401341_kernel<<<grid, block, 0, stream>>>(inp, sw, out);
}